// TriPlaneGenerator_74534862455178
// MI455X (gfx1250) — compile-verified
//
#include <hip/hip_runtime.h>

// Tri-plane bilinear sampling for MI455X (gfx1250, wave32).
//
// plane_features: (1, 3, 32, 256, 256) f32   coordinates: (1, M, 3) f32
// output:         (1, 3, M, 32) f32          M = 1048576
//
// Strategy:
//  1) One-time transpose (C,H,W) -> (H,W,C) into d_ws so each pixel's 32
//     channels are one 128B cacheline (24 MB total, stays in the 192 MB L2).
//     The transpose stages tiles through LDS using CDNA5 async global<->LDS
//     DMA (ASYNCcnt) when the toolchain exposes the builtins.
//  2) Sampling kernel: lane = channel (wave32 == C == 32). Each bilinear
//     corner gather is a single fully-coalesced 128B request per wave;
//     output stores are coalesced 128B per wave, written with NT hints so the
//     384 MB output stream does not evict the L2-resident feature planes.

namespace {
constexpr int C  = 32;
constexpr int Wd = 256;
constexpr int Hd = 256;
constexpr int HW = Wd * Hd;
constexpr int NP = 3;
constexpr int PTS_PER_WAVE = 4;
}

#if defined(__AMDGCN__) &&                                                     \
    __has_builtin(__builtin_amdgcn_global_load_async_to_lds_b32) &&            \
    __has_builtin(__builtin_amdgcn_global_store_async_from_lds_b32) &&         \
    __has_builtin(__builtin_amdgcn_s_wait_asynccnt)
#define TRIPLANE_ASYNC_LDS 1
#else
#define TRIPLANE_ASYNC_LDS 0
#endif

#if TRIPLANE_ASYNC_LDS
// b32 variants take int* parameters (per hipcc diagnostic).
#define GPTR_I32(p) ((__attribute__((address_space(1))) int*)(p))
#define LPTR_I32(p) ((__attribute__((address_space(3))) int*)(p))
#endif

// ---------------------------------------------------------------------------
// (P,C,H,W) -> (P,H*W,C) transpose, LDS-tiled 32x32, conflict-free padding.
// grid.x = P * (HW/32); 256 threads per block (8 wave32).
// ---------------------------------------------------------------------------
__global__ __launch_bounds__(256) void triplane_transpose_kernel(
    const float* __restrict__ src, float* __restrict__ dst)
{
  __shared__ float tile[C][C + 1];   // +1 pad: stride 33 -> conflict-free

  const int tid  = threadIdx.x;
  const int lane = tid & 31;
  const int row8 = (tid >> 5) & 7;               // provably 0..7

  const int tilesPerPlane = HW / 32;             // 2048
  const int p   = blockIdx.x / tilesPerPlane;    // 0..2
  const int t   = blockIdx.x % tilesPerPlane;
  const long long hw0 = (long long)t * 32;

  const float* in = src + (long long)p * C * HW + hw0;

#if TRIPLANE_ASYNC_LDS
  // CDNA5 async DMA: global -> LDS directly, no VGPR round trip, one
  // s_wait_asynccnt for the whole batch of 4 coalesced 128B rows.
#pragma unroll
  for (int i = 0; i < 4; ++i) {
    const int cc = row8 + 8 * i;
    __builtin_amdgcn_global_load_async_to_lds_b32(
        GPTR_I32(in + (long long)cc * HW + lane),
        LPTR_I32(&tile[cc][lane]),
        /*offset=*/0, /*cpol=*/0);
  }
  __builtin_amdgcn_s_wait_asynccnt(0);
#else
#pragma unroll
  for (int i = 0; i < 4; ++i) {
    const int cc = row8 + 8 * i;
    tile[cc][lane] = in[(long long)cc * HW + lane];
  }
#endif

  __syncthreads();

  float* o = dst + ((long long)p * HW + hw0) * C;

#if TRIPLANE_ASYNC_LDS
  // LDS -> global async store (coalesced 128B rows); S_ENDPGM waits idle.
#pragma unroll
  for (int i = 0; i < 4; ++i) {
    const int rr = row8 + 8 * i;
    __builtin_amdgcn_global_store_async_from_lds_b32(
        GPTR_I32(o + (long long)rr * C + lane),
        LPTR_I32(&tile[lane][rr]),
        /*offset=*/0, /*cpol=*/0);
  }
  __builtin_amdgcn_s_wait_asynccnt(0);
#else
#pragma unroll
  for (int i = 0; i < 4; ++i) {
    const int rr = row8 + 8 * i;
    o[(long long)rr * C + lane] = tile[lane][rr];
  }
#endif
}

// ---------------------------------------------------------------------------
// One point (all 3 planes) sampled by one wave; lane == channel.
// ---------------------------------------------------------------------------
template <bool TRANSPOSED>
__device__ __forceinline__ void sample_point(
    const float* __restrict__ feats, float* __restrict__ out, int M,
    long long m, int lane, float cx, float cy, float cz)
{
  // Plane matrices are permutations; 2/box_warp == 1:
  // proj0 = (x,y), proj1 = (y,z), proj2 = (z,x).
  const float us[NP] = {cx, cy, cz};
  const float vs[NP] = {cy, cz, cx};

#pragma unroll
  for (int p = 0; p < NP; ++p) {
    // align_corners=False: ix = (u+1)*W/2 - 0.5 = u*128 + 127.5
    const float ix  = fmaf(us[p], 128.0f, 127.5f);
    const float iy  = fmaf(vs[p], 128.0f, 127.5f);
    const float fx0 = floorf(ix);
    const float fy0 = floorf(iy);
    const float wx1 = ix - fx0, wy1 = iy - fy0;
    const float wx0 = 1.0f - wx1, wy0 = 1.0f - wy1;
    const int   x0  = (int)fx0, y0 = (int)fy0;

    float r;
    // Wave-uniform branch: every lane shares the same sample point.
    if (x0 >= 0 && x0 < Wd - 1 && y0 >= 0 && y0 < Hd - 1) {
      if (TRANSPOSED) {
        // 4 corners, each one coalesced 128B wave load from L2.
        const float* b =
            feats + ((long long)p * HW + (long long)y0 * Wd + x0) * C + lane;
        const float v00 = b[0];
        const float v01 = b[C];
        const float v10 = b[Wd * C];
        const float v11 = b[Wd * C + C];
        r = (v00 * wx0 + v01 * wx1) * wy0 + (v10 * wx0 + v11 * wx1) * wy1;
      } else {
        const float* b =
            feats + ((long long)p * C + lane) * HW + (long long)y0 * Wd + x0;
        const float v00 = b[0];
        const float v01 = b[1];
        const float v10 = b[Wd];
        const float v11 = b[Wd + 1];
        r = (v00 * wx0 + v01 * wx1) * wy0 + (v10 * wx0 + v11 * wx1) * wy1;
      }
    } else {
      // Border: per-corner zero-padding, still wave-uniform.
      r = 0.0f;
#pragma unroll
      for (int dy = 0; dy < 2; ++dy) {
#pragma unroll
        for (int dx = 0; dx < 2; ++dx) {
          const int xx = x0 + dx, yy = y0 + dy;
          if (xx >= 0 && xx < Wd && yy >= 0 && yy < Hd) {
            const float w = (dx ? wx1 : wx0) * (dy ? wy1 : wy0);
            const float v = TRANSPOSED
                ? feats[((long long)p * HW + (long long)yy * Wd + xx) * C + lane]
                : feats[((long long)p * C + lane) * HW + (long long)yy * Wd + xx];
            r = fmaf(v, w, r);
          }
        }
      }
    }

    // Coalesced 128B store per wave; NT hint keeps the 384MB output stream
    // from evicting the L2-resident feature planes.
    __builtin_nontemporal_store(r, out + ((long long)p * M + m) * C + lane);
  }
}

// ---------------------------------------------------------------------------
// Sampling kernel. One wave handles PTS_PER_WAVE points x 3 planes.
// ---------------------------------------------------------------------------
template <bool TRANSPOSED>
__global__ __launch_bounds__(256) void triplane_sample_kernel(
    const float* __restrict__ feats,
    const float* __restrict__ coords,
    float* __restrict__ out,
    int M)
{
  const int lane = threadIdx.x & 31;
  const long long wave =
      (long long)blockIdx.x * (blockDim.x >> 5) + (threadIdx.x >> 5);
  const long long m0 = wave * PTS_PER_WAVE;
  if (m0 >= (long long)M) return;

  if (m0 + PTS_PER_WAVE <= (long long)M) {
    // Fast path: issue all 12 coordinate loads up front (MLP), then sample.
    float c3[PTS_PER_WAVE][3];
#pragma unroll
    for (int k = 0; k < PTS_PER_WAVE; ++k) {
      const float* cp = coords + (m0 + k) * 3;
      c3[k][0] = __builtin_nontemporal_load(cp + 0);
      c3[k][1] = __builtin_nontemporal_load(cp + 1);
      c3[k][2] = __builtin_nontemporal_load(cp + 2);
    }
#pragma unroll
    for (int k = 0; k < PTS_PER_WAVE; ++k)
      sample_point<TRANSPOSED>(feats, out, M, m0 + k, lane,
                               c3[k][0], c3[k][1], c3[k][2]);
  } else {
    // Tail (only when M % PTS_PER_WAVE != 0).
    for (long long m = m0; m < (long long)M; ++m) {
      const float* cp = coords + m * 3;
      const float cx = __builtin_nontemporal_load(cp + 0);
      const float cy = __builtin_nontemporal_load(cp + 1);
      const float cz = __builtin_nontemporal_load(cp + 2);
      sample_point<TRANSPOSED>(feats, out, M, m, lane, cx, cy, cz);
    }
  }
}

// ---------------------------------------------------------------------------
extern "C" void kernel_launch(void* const* d_in, const int* in_sizes, int n_in,
                              void* d_out, int out_size, void* d_ws, size_t ws_size,
                              hipStream_t stream) {
  const float* planes = (const float*)d_in[0];  // (1,3,32,256,256)
  const float* coords = (const float*)d_in[1];  // (1,M,3)
  float* out = (float*)d_out;                   // (1,3,M,32)

  const int M = in_sizes[1] / 3;

  const long long waves  = ((long long)M + PTS_PER_WAVE - 1) / PTS_PER_WAVE;
  const long long blocks = (waves + 7) / 8;     // 8 wave32 per 256-thread block

  const size_t needed = (size_t)NP * HW * C * sizeof(float);  // 24 MB
  if (d_ws != nullptr && ws_size >= needed) {
    float* tp = (float*)d_ws;
    triplane_transpose_kernel<<<dim3(NP * (HW / 32)), dim3(256), 0, stream>>>(
        planes, tp);
    triplane_sample_kernel<true><<<dim3((unsigned)blocks), dim3(256), 0, stream>>>(
        tp, coords, out, M);
  } else {
    triplane_sample_kernel<false><<<dim3((unsigned)blocks), dim3(256), 0, stream>>>(
        planes, coords, out, M);
  }
}